// LiteLinear_30975304138921
// MI455X (gfx1250) — compile-verified
//
#include <hip/hip_runtime.h>

typedef __attribute__((ext_vector_type(2))) float v2f;
typedef __attribute__((ext_vector_type(4))) float v4f;
typedef __attribute__((ext_vector_type(8))) float v8f;
typedef __attribute__((ext_vector_type(4))) int   i4;

#define D_IN     4096
#define D_OUT    4096
#define N_TOK    8192
#define RANK     16
#define NADP     8

#if __has_builtin(__builtin_amdgcn_global_load_async_to_lds_b128) && \
    __has_builtin(__builtin_amdgcn_s_wait_asynccnt)
#define USE_ASYNC_LDS 1
#endif

#define GAS __attribute__((address_space(1)))
#define LAS __attribute__((address_space(3)))

// -------------------- Kernel 1: LoRA down-projection --------------------
// h[t][r] = scaling[id-1] * dot(x[t], lora_a[id-1][r]); zeros when id==0.
__global__ __launch_bounds__(256) void lora_down_kernel(
    const float* __restrict__ x, const float* __restrict__ lora_a,
    const float* __restrict__ scalings, const int* __restrict__ mapping,
    float* __restrict__ h)
{
    __shared__ float xs[D_IN];
    const int t   = blockIdx.x;
    const int tid = threadIdx.x;
    const int id  = mapping[t];                 // block-uniform
    if (id == 0) {
        if (tid < RANK) h[t * RANK + tid] = 0.0f;
        return;
    }
    const float* xrow = x + (size_t)t * D_IN;
    for (int k = tid * 4; k < D_IN; k += 256 * 4)
        *(v4f*)&xs[k] = *(const v4f*)&xrow[k];
    __syncthreads();

    const int wave = tid >> 5;
    const int lane = tid & 31;
    const float* A = lora_a + (size_t)(id - 1) * RANK * D_IN;
    const float  s = scalings[id - 1];

    #pragma unroll
    for (int rr = 0; rr < 2; ++rr) {
        const int r = wave * 2 + rr;
        const float* arow = A + (size_t)r * D_IN;
        float acc = 0.0f;
        for (int k = lane; k < D_IN; k += 32)
            acc += xs[k] * arow[k];
        #pragma unroll
        for (int off = 16; off >= 1; off >>= 1)
            acc += __shfl_xor(acc, off, 32);
        if (lane == 0) h[t * RANK + r] = acc * s;
    }
}

// -------------------- Kernel 2: base GEMM + bias (fp32 WMMA) --------------------
#define BM  128
#define BN  128
#define BK  32
#define LDT 36   // padded LDS row stride (floats); 144 B keeps 16B/8B alignment
#define NCH (D_IN / BK)

// Stage one 128x32 x-tile and one 128x32 W-tile into LDS buffers.
// Async path: data goes straight HBM->LDS (ASYNCcnt), no VGPR round-trip.
__device__ __forceinline__ void stage_tiles(
    const float* __restrict__ x, const float* __restrict__ w,
    float* __restrict__ Asb, float* __restrict__ Bsb,
    int m0, int n0, int k0, int tid)
{
    #pragma unroll
    for (int it = 0; it < 4; ++it) {
        const int j   = tid + it * 256;
        const int row = j >> 3;
        const int c4  = (j & 7) << 2;
        const float* ga = &x[(size_t)(m0 + row) * D_IN + k0 + c4];
        const float* gb = &w[(size_t)(n0 + row) * D_IN + k0 + c4];
        float* la = &Asb[row * LDT + c4];
        float* lb = &Bsb[row * LDT + c4];
#ifdef USE_ASYNC_LDS
        __builtin_amdgcn_global_load_async_to_lds_b128(
            (GAS i4*)ga, (LAS i4*)la, 0, 0);
        __builtin_amdgcn_global_load_async_to_lds_b128(
            (GAS i4*)gb, (LAS i4*)lb, 0, 0);
#else
        *(v4f*)la = *(const v4f*)ga;
        *(v4f*)lb = *(const v4f*)gb;
#endif
    }
}

__global__ __launch_bounds__(256) void base_gemm_kernel(
    const float* __restrict__ x, const float* __restrict__ w,
    const float* __restrict__ bias, float* __restrict__ out)
{
    __shared__ float As[2][BM * LDT];   // double-buffered x tiles
    __shared__ float Bs[2][BN * LDT];   // double-buffered W tiles

    const int tid  = threadIdx.x;
    const int lane = tid & 31;
    const int wave = tid >> 5;
    const int wm   = wave >> 2;          // 0..1 : 64-row slab
    const int wn   = wave & 3;           // 0..3 : 32-col slab
    const int m0   = blockIdx.y * BM;
    const int n0   = blockIdx.x * BN;

    const int half = lane >> 4;          // which K-pair this lane supplies
    const int l15  = lane & 15;

    v8f c[4][2] = {};

    // prologue: chunk 0 in flight
    stage_tiles(x, w, As[0], Bs[0], m0, n0, 0, tid);

    for (int ch = 0; ch < NCH; ++ch) {
        const int cur = ch & 1;
        if (ch + 1 < NCH) {
            // issue next chunk into the other buffer, then wait for current
            stage_tiles(x, w, As[cur ^ 1], Bs[cur ^ 1], m0, n0, (ch + 1) * BK, tid);
#ifdef USE_ASYNC_LDS
            __builtin_amdgcn_s_wait_asynccnt(8);   // in-order: first 8 (chunk ch) done
#endif
        } else {
#ifdef USE_ASYNC_LDS
            __builtin_amdgcn_s_wait_asynccnt(0);
#endif
        }
        __syncthreads();

        const float* Asb = As[cur];
        const float* Bsb = Bs[cur];
        #pragma unroll
        for (int kk = 0; kk < BK; kk += 4) {
            v2f a[4], b[2];
            #pragma unroll
            for (int mt = 0; mt < 4; ++mt) {
                const int mrow = wm * 64 + mt * 16 + l15;
                a[mt] = *(const v2f*)&Asb[mrow * LDT + kk + half * 2];
            }
            #pragma unroll
            for (int nt = 0; nt < 2; ++nt) {
                const int ncol = wn * 32 + nt * 16 + l15;
                b[nt] = *(const v2f*)&Bsb[ncol * LDT + kk + half * 2];
            }
            #pragma unroll
            for (int mt = 0; mt < 4; ++mt)
                #pragma unroll
                for (int nt = 0; nt < 2; ++nt)
                    c[mt][nt] = __builtin_amdgcn_wmma_f32_16x16x4_f32(
                        false, a[mt], false, b[nt],
                        (short)0, c[mt][nt], false, false);
        }
        __syncthreads();
    }

    // epilogue: C layout -> row = base + half*8 + v, col = base + l15
    #pragma unroll
    for (int nt = 0; nt < 2; ++nt) {
        const int col = n0 + wn * 32 + nt * 16 + l15;
        const float bz = bias[col];
        #pragma unroll
        for (int mt = 0; mt < 4; ++mt) {
            const int rbase = m0 + wm * 64 + mt * 16 + half * 8;
            #pragma unroll
            for (int v = 0; v < 8; ++v)
                out[(size_t)(rbase + v) * D_OUT + col] = c[mt][nt][v] + bz;
        }
    }
}

// -------------------- Kernel 3: LoRA up-projection scatter-add --------------------
__device__ __forceinline__ float dot4(v4f a, v4f b) {
    return a.x * b.x + a.y * b.y + a.z * b.z + a.w * b.w;
}

__global__ __launch_bounds__(256) void lora_up_kernel(
    const float* __restrict__ h, const float* __restrict__ lora_b,
    const int* __restrict__ mapping, float* __restrict__ out)
{
    const int t  = blockIdx.x;
    const int id = mapping[t];                  // block-uniform
    if (id == 0) return;

    const float* hrow = h + (size_t)t * RANK;
    const v4f h0 = *(const v4f*)&hrow[0];
    const v4f h1 = *(const v4f*)&hrow[4];
    const v4f h2 = *(const v4f*)&hrow[8];
    const v4f h3 = *(const v4f*)&hrow[12];

    const float* B = lora_b + (size_t)(id - 1) * (size_t)D_OUT * RANK;
    float* orow    = out + (size_t)t * D_OUT;
    const int tid  = threadIdx.x;

    #pragma unroll
    for (int i = 0; i < D_OUT / 256; ++i) {
        const int col = tid + i * 256;          // coalesced across lanes
        const float* br = B + (size_t)col * RANK;
        float acc = dot4(h0, *(const v4f*)&br[0])
                  + dot4(h1, *(const v4f*)&br[4])
                  + dot4(h2, *(const v4f*)&br[8])
                  + dot4(h3, *(const v4f*)&br[12]);
        orow[col] += acc;
    }
}

// -------------------- launch --------------------
extern "C" void kernel_launch(void* const* d_in, const int* in_sizes, int n_in,
                              void* d_out, int out_size, void* d_ws, size_t ws_size,
                              hipStream_t stream) {
    const float* x        = (const float*)d_in[0];
    const float* weight   = (const float*)d_in[1];
    const float* bias     = (const float*)d_in[2];
    const float* lora_a   = (const float*)d_in[3];
    const float* lora_b   = (const float*)d_in[4];
    const float* scalings = (const float*)d_in[5];
    const int*   mapping  = (const int*)d_in[6];
    float* out = (float*)d_out;
    float* h   = (float*)d_ws;                 // N_TOK * RANK floats (512 KB)

    lora_down_kernel<<<N_TOK, 256, 0, stream>>>(x, lora_a, scalings, mapping, h);
    base_gemm_kernel<<<dim3(D_OUT / BN, N_TOK / BM), 256, 0, stream>>>(x, weight, bias, out);
    lora_up_kernel<<<N_TOK, 256, 0, stream>>>(h, lora_b, mapping, out);
}